// ProposalRefine_61856118997269
// MI455X (gfx1250) — compile-verified
//
#include <hip/hip_runtime.h>
#include <hip/hip_bf16.h>

// ---------------------------------------------------------------------------
// Problem constants (from the reference)
// ---------------------------------------------------------------------------
#define NIMG 16
#define NA   32768
#define NB   32
#define NTOT (NA + NB)      // 32800 candidate boxes per image
#define NSEL 256
#define NFG  64
#define NBG  192
#define NCLS 21

typedef float v2f __attribute__((ext_vector_type(2)));
typedef float v8f __attribute__((ext_vector_type(8)));
typedef int   v4i_vs __attribute__((__vector_size__(16)));  // matches builtin param

#define GLOBAL_AS __attribute__((address_space(1)))
#define LDS_AS    __attribute__((address_space(3)))

// ---------------------------------------------------------------------------
// CDNA5 async global->LDS staging (guarded; falls back to plain copies)
// ---------------------------------------------------------------------------
#if defined(__gfx1250__) && __has_builtin(__builtin_amdgcn_global_load_async_to_lds_b128)
#define HAVE_ASYNC_B128 1
#endif
#if defined(__gfx1250__) && __has_builtin(__builtin_amdgcn_global_load_async_to_lds_b32)
#define HAVE_ASYNC_B32 1
#endif

__device__ __forceinline__ void async_gl2lds_b128(const void* g, void* l) {
#if defined(HAVE_ASYNC_B128)
  __builtin_amdgcn_global_load_async_to_lds_b128(
      (GLOBAL_AS v4i_vs*)g, (LDS_AS v4i_vs*)l, 0, 0);
#else
  *(float4*)l = *(const float4*)g;
#endif
}

__device__ __forceinline__ void async_gl2lds_b32(const void* g, void* l) {
#if defined(HAVE_ASYNC_B32)
  __builtin_amdgcn_global_load_async_to_lds_b32(
      (GLOBAL_AS int*)g, (LDS_AS int*)l, 0, 0);
#else
  *(int*)l = *(const int*)g;
#endif
}

__device__ __forceinline__ void wait_async_zero() {
#if defined(HAVE_ASYNC_B128) || defined(HAVE_ASYNC_B32)
#if __has_builtin(__builtin_amdgcn_s_wait_asynccnt)
  __builtin_amdgcn_s_wait_asynccnt(0);
#else
  asm volatile("s_wait_asynccnt 0" ::: "memory");
#endif
#endif
}

__device__ __forceinline__ unsigned ballot32(bool p) {
#if __has_builtin(__builtin_amdgcn_ballot_w32)
  return __builtin_amdgcn_ballot_w32(p);
#else
  return (unsigned)__ballot(p);
#endif
}

// ---------------------------------------------------------------------------
// Threefry2x32-20 (JAX default PRNG) — deterministic sampling to mirror ref
// ---------------------------------------------------------------------------
__device__ __forceinline__ unsigned rotl32(unsigned x, int r) {
  return (x << r) | (x >> (32 - r));
}

__device__ __forceinline__ void threefry2x32(unsigned k0, unsigned k1,
                                             unsigned c0, unsigned c1,
                                             unsigned& o0, unsigned& o1) {
  const unsigned ks2 = 0x1BD11BDAu ^ k0 ^ k1;
  unsigned x0 = c0 + k0, x1 = c1 + k1;
  const int rotA[4] = {13, 15, 26, 6};
  const int rotB[4] = {17, 29, 16, 24};
#pragma unroll
  for (int g = 0; g < 5; ++g) {
#pragma unroll
    for (int r = 0; r < 4; ++r) {
      const int rr = (g & 1) ? rotB[r] : rotA[r];
      x0 += x1;
      x1 = rotl32(x1, rr);
      x1 ^= x0;
    }
    unsigned i0, i1;
    switch (g) {
      case 0:  i0 = k1;  i1 = ks2; break;
      case 1:  i0 = ks2; i1 = k0;  break;
      case 2:  i0 = k0;  i1 = k1;  break;
      case 3:  i0 = k1;  i1 = ks2; break;
      default: i0 = ks2; i1 = k0;  break;
    }
    x0 += i0;
    x1 += i1 + (unsigned)(g + 1);
  }
  o0 = x0;
  o1 = x1;
}

// random_bits(key(1234), iota(64))[i] per JAX's half-split counter layout
__device__ __forceinline__ unsigned jax_root_bits64(int i) {
  const int p = (i < 32) ? i : i - 32;
  unsigned x, y;
  threefry2x32(0u, 1234u, (unsigned)p, (unsigned)(p + 32), x, y);
  return (i < 32) ? x : y;
}

// uniform(key,(k,))[p] with k even
__device__ __forceinline__ float jax_uniform(unsigned k0, unsigned k1, int k, int p) {
  const int half = k >> 1;
  const int pp = p % half;
  const int which = p / half;
  unsigned x, y;
  threefry2x32(k0, k1, (unsigned)pp, (unsigned)(pp + half), x, y);
  const unsigned bits = which ? y : x;
  return __uint_as_float((bits >> 9) | 0x3f800000u) - 1.0f;
}

// ---------------------------------------------------------------------------
// Kernel A: stream 32800 ROIs/image, classify vs 32 LDS-staged GT boxes
//   codes[n*NTOT+r] = (first_fg_gt << 8) | {fg=1, bg=2, backup=4}
// ---------------------------------------------------------------------------
#define KA_ITER 4

__global__ void kA_classify(const float* __restrict__ rois,
                            const float* __restrict__ gtb,
                            int* __restrict__ codes) {
  __shared__ float4 sgt[NB];
  const int n = blockIdx.y;
  const int tid = threadIdx.x;

  if (tid < NB) {
    async_gl2lds_b128(gtb + ((size_t)n * NB + tid) * 4, &sgt[tid]);
  }
  wait_async_zero();
  __syncthreads();

  const int base = blockIdx.x * (blockDim.x * KA_ITER);
#pragma unroll
  for (int i = 0; i < KA_ITER; ++i) {
    const int r = base + i * 256 + tid;
    if (r >= NTOT) continue;
    // prefetch next iteration's cache line (global_prefetch_b8)
    const int rn = r + 256;
    if (i + 1 < KA_ITER && rn < NA)
      __builtin_prefetch(rois + ((size_t)n * NA + rn) * 4, 0, 1);

    float4 bx;
    if (r < NA)
      bx = *(const float4*)(rois + ((size_t)n * NA + r) * 4);
    else
      bx = sgt[r - NA];

    const float areaB = (bx.z - bx.x + 1.f) * (bx.w - bx.y + 1.f);
    int code = 0, fgidx = 0;
    bool fgfound = false;
#pragma unroll
    for (int g = 0; g < NB; ++g) {
      const float4 gb = sgt[g];
      const float ix1 = fmaxf(bx.x, gb.x);
      const float iy1 = fmaxf(bx.y, gb.y);
      const float ix2 = fminf(bx.z, gb.z);
      const float iy2 = fminf(bx.w, gb.w);
      const float iw = fmaxf(ix2 - ix1 + 1.f, 0.f);
      const float ih = fmaxf(iy2 - iy1 + 1.f, 0.f);
      const float inter = iw * ih;
      const float areaG = (gb.z - gb.x + 1.f) * (gb.w - gb.y + 1.f);
      const float ov = inter / (areaB + areaG - inter);
      if (ov >= 0.5f) {
        code |= 1;
        if (!fgfound) { fgidx = g; fgfound = true; }
      }
      if (ov >= 0.1f && ov < 0.5f) code |= 2;
      if (ov >= 0.0f && ov < 0.1f) code |= 4;
    }
    codes[(size_t)n * NTOT + r] = code | (fgidx << 8);
  }
}

// ---------------------------------------------------------------------------
// Kernel B: ordered compaction. One wave32 per (image, list-type).
//   lists[(ty*NIMG+n)*NTOT + pos] = r (ascending), cnts[n*4+ty] = count
// ---------------------------------------------------------------------------
__global__ void kB_compact(const int* __restrict__ codes,
                           int* __restrict__ lists,
                           int* __restrict__ cnts) {
  const int n = blockIdx.x;
  const int ty = blockIdx.y;  // 0=fg 1=bg 2=backup
  const int lane = threadIdx.x;
  int* list = lists + ((size_t)ty * NIMG + n) * NTOT;
  const int bit = 1 << ty;
  const unsigned lt = (1u << lane) - 1u;
  int cnt = 0;
  for (int c = 0; c < NTOT; c += 32) {
    const int r = c + lane;
    const bool pred = (r < NTOT) && ((codes[(size_t)n * NTOT + r] & bit) != 0);
    const unsigned mask = ballot32(pred);
    if (pred) list[cnt + __popc(mask & lt)] = r;
    cnt += __popc(mask);
  }
  if (lane == 0) cnts[n * 4 + ty] = cnt;
}

// ---------------------------------------------------------------------------
// Kernel C: threefry sampling + gathers + bbox coefficients.
// GT box/class gather is a one-hot matmul on the matrix pipe:
//   out(16x16 tile) += onehot(16x4) @ gt(4x16)   via V_WMMA_F32_16X16X4_F32
// B operand is pre-materialized in a padded LDS table so the K-loop is pure
// v_cmp/v_cndmask + v_wmma (no divergent ds loads between WMMAs).
// ---------------------------------------------------------------------------
#define BSTRIDE 34   // 32 + pad, spreads LDS banks, keeps 8B alignment
#define TSTRIDE 17   // 16 + pad for stile rows

__global__ void kC_sample(const float* __restrict__ rois,
                          const float* __restrict__ gtb,
                          const int* __restrict__ gtc,
                          const int* __restrict__ codes,
                          const int* __restrict__ lists,
                          const int* __restrict__ cnts,
                          float* __restrict__ outBoxes,
                          float* __restrict__ outLabels,
                          float* __restrict__ outCoef) {
  __shared__ float4 sgt[NB];
  __shared__ int sclsi[NB];
  __shared__ float scls[NB];
  __shared__ int gtm[NSEL];
  __shared__ float btab[16][BSTRIDE];      // B matrix: [col][k]
  __shared__ float stile[16 * 16 * TSTRIDE];  // 16 tiles of 16x16 (padded rows)

  const int n = blockIdx.x;
  const int t = threadIdx.x;

  // --- stage GT boxes (b128) and GT classes (b32) via async-to-LDS ---
  if (t < NB) {
    async_gl2lds_b128(gtb + ((size_t)n * NB + t) * 4, &sgt[t]);
  } else if (t >= 32 && t < 32 + NB) {
    const int g = t - 32;
    async_gl2lds_b32(gtc + (size_t)n * NB + g, &sclsi[g]);
  }
  wait_async_zero();
  __syncthreads();
  if (t < NB) scls[t] = (float)sclsi[t];
  __syncthreads();

  // --- build B table once: btab[c][k] = c<4 ? gtbox[k][c] : c==4 ? cls[k] : 0
#pragma unroll
  for (int idx = t; idx < 16 * NB; idx += NSEL) {
    const int c = idx >> 5;       // 0..15
    const int k = idx & (NB - 1); // 0..31
    float v = 0.f;
    if (c < 4) {
      const float4 gb = sgt[k];
      v = (c == 0) ? gb.x : (c == 1) ? gb.y : (c == 2) ? gb.z : gb.w;
    } else if (c == 4) {
      v = scls[k];
    }
    btab[c][k] = v;
  }

  // --- per-slot sampling (matches reference _pick semantics) ---
  const int fgCnt = cnts[n * 4 + 0];
  const int bgCnt = cnts[n * 4 + 1];
  const int bkCnt = cnts[n * 4 + 2];
  const bool isfg = (t < NFG);

  int num, kTot, p, d;
  const int* list;
  if (isfg) {
    num = fgCnt; kTot = NFG; p = t;
    list = lists + ((size_t)0 * NIMG + n) * NTOT;
    d = 2 * n;
  } else {
    const bool useBg = (bgCnt > 0);
    num = useBg ? bgCnt : bkCnt;
    kTot = NBG; p = t - NFG;
    list = lists + ((size_t)(useBg ? 1 : 2) * NIMG + n) * NTOT;
    d = 2 * n + 1;
  }
  // derive JAX split key #d of key(1234)
  const unsigned k0 = jax_root_bits64(2 * d);
  const unsigned k1 = jax_root_bits64(2 * d + 1);
  const float u = jax_uniform(k0, k1, kTot, p);
  const int numc = (num > 1) ? num : 1;
  int rnd = (int)(u * (float)numc);
  rnd = (rnd < numc - 1) ? rnd : (numc - 1);
  const int fill = (p < num) ? p : rnd;
  const int sel = (num > 0) ? list[fill] : (NTOT - 1);  // OOB clamp like JAX

  int gm = 0;
  if (isfg) gm = (codes[(size_t)n * NTOT + sel] >> 8) & (NB - 1);
  gtm[t] = gm;

  // selected ROI box -> output + kept in registers
  float4 rb;
  if (sel < NA)
    rb = *(const float4*)(rois + ((size_t)n * NA + sel) * 4);
  else
    rb = sgt[sel - NA];
  {
    float* ob = outBoxes + ((size_t)n * NSEL + t) * 4;
    ob[0] = rb.x; ob[1] = rb.y; ob[2] = rb.z; ob[3] = rb.w;
  }
  __syncthreads();

  // --- one-hot gather of matched GT row [x1,y1,x2,y2,class] ---
  float4 g;
  float labf;
#if defined(__gfx1250__) && __has_builtin(__builtin_amdgcn_wmma_f32_16x16x4_f32)
  {
    const int lane = t & 31;
    const int w = t >> 5;       // wave id (8 waves)
    const int m = lane & 15;    // A row / B,C,D column within tile
    const int hh = lane >> 4;   // K-half selector (per 16x16x4 f32 layout)

    // B fragments are tile-invariant: hoist all 8 (one ds_load_b64 each)
    v2f bfrag[8];
#pragma unroll
    for (int s = 0; s < 8; ++s)
      bfrag[s] = *(const v2f*)&btab[m][4 * s + 2 * hh];

#pragma unroll
    for (int tile = 0; tile < 2; ++tile) {
      const int R = w * 32 + tile * 16;  // first output row of this tile
      const int gmr = gtm[R + m];        // hoisted out of the K-loop
      v8f acc = {0.f, 0.f, 0.f, 0.f, 0.f, 0.f, 0.f, 0.f};
#pragma unroll
      for (int s = 0; s < 8; ++s) {      // K = 32 in steps of 4
        const int kb = 4 * s + 2 * hh;
        v2f a;
        a.x = (gmr == kb) ? 1.f : 0.f;
        a.y = (gmr == kb + 1) ? 1.f : 0.f;
        acc = __builtin_amdgcn_wmma_f32_16x16x4_f32(
            false, a, false, bfrag[s], (short)0, acc, false, false);
      }
      float* reg = stile + (size_t)(w * 2 + tile) * (16 * TSTRIDE);
#pragma unroll
      for (int v = 0; v < 8; ++v) reg[(hh * 8 + v) * TSTRIDE + m] = acc[v];
    }
    __syncthreads();
    const float* reg = stile + (size_t)(t >> 4) * (16 * TSTRIDE);
    const int row = t & 15;
    g.x = reg[row * TSTRIDE + 0];
    g.y = reg[row * TSTRIDE + 1];
    g.z = reg[row * TSTRIDE + 2];
    g.w = reg[row * TSTRIDE + 3];
    labf = reg[row * TSTRIDE + 4];
  }
#else
  g = sgt[gm];
  labf = scls[gm];
  __syncthreads();
#endif

  const int lab = isfg ? (int)(labf + 0.5f) : 0;

  // --- bbox regression coefficients ---
  const float bw = rb.z - rb.x + 1.f;
  const float bh = rb.w - rb.y + 1.f;
  const float cx = rb.x + 0.5f * bw;
  const float cy = rb.y + 0.5f * bh;
  const float gw = g.z - g.x + 1.f;
  const float gh = g.w - g.y + 1.f;
  const float gcx = g.x + 0.5f * gw;
  const float gcy = g.y + 0.5f * gh;
  const float c0 = (gcx - cx) / bw;
  const float c1 = (gcy - cy) / bh;
  const float c2 = logf(gw / bw);
  const float c3 = logf(gh / bh);

  outLabels[(size_t)n * NSEL + t] = (float)lab;

  float* crow = outCoef + ((size_t)n * NSEL + t) * (NCLS * 4);
#pragma unroll 4
  for (int j = 0; j < NCLS * 4; ++j) crow[j] = 0.f;
  if (isfg) {
    const int o = lab * 4;
    crow[o + 0] = c0;
    crow[o + 1] = c1;
    crow[o + 2] = c2;
    crow[o + 3] = c3;
  }
}

// ---------------------------------------------------------------------------
// Host launcher
// ---------------------------------------------------------------------------
extern "C" void kernel_launch(void* const* d_in, const int* in_sizes, int n_in,
                              void* d_out, int out_size, void* d_ws, size_t ws_size,
                              hipStream_t stream) {
  (void)in_sizes; (void)n_in; (void)out_size; (void)ws_size;
  const float* rois = (const float*)d_in[0];
  const float* gtb  = (const float*)d_in[1];
  const int*   gtc  = (const int*)d_in[2];

  float* out = (float*)d_out;
  float* outBoxes  = out;                              // 16*256*4
  float* outLabels = out + (size_t)NIMG * NSEL * 4;    // 16*256
  float* outCoef   = outLabels + (size_t)NIMG * NSEL;  // 16*256*84

  char* ws = (char*)d_ws;
  const size_t codesBytes = (size_t)NIMG * NTOT * sizeof(int);
  int* codes = (int*)ws;                               // [N][NTOT]
  int* lists = (int*)(ws + codesBytes);                // [3][N][NTOT]
  int* cnts  = (int*)(ws + codesBytes * 4);            // [N][4]

  const int kaBlocksX = (NTOT + 256 * KA_ITER - 1) / (256 * KA_ITER);  // 33
  kA_classify<<<dim3(kaBlocksX, NIMG), 256, 0, stream>>>(rois, gtb, codes);
  kB_compact<<<dim3(NIMG, 3), 32, 0, stream>>>(codes, lists, cnts);
  kC_sample<<<NIMG, 256, 0, stream>>>(rois, gtb, gtc, codes, lists, cnts,
                                      outBoxes, outLabels, outCoef);
}